// Patch_Embed_Center_Rotate_64433099374842
// MI455X (gfx1250) — compile-verified
//
#include <hip/hip_runtime.h>
#include <cstdint>

// ---------------------------------------------------------------------------
// Patch_Embed_Center_Rotate for x:(256,3,224,224) fp32, P=16, grid 14x14.
// Semantics (derived from the JAX reference):
//   identity copy everywhere, EXCEPT patches with grid col j in {5..9} and
//   grid row k in {5..9} (pixel window rows/cols [80,160)), which are
//   transposed within the 16x16 patch.
//
// Zero-FLOP op -> pure HBM roofline: 308 MiB traffic / 23.3 TB/s ~= 13.3 us.
// Strategy: b128 NT streaming copy for the 87% identity region; CDNA5 async
// global->LDS b128 (ASYNCcnt + s_wait_asynccnt) staged transpose for the 13%
// center region so reads AND writes stay 128b-coalesced.
// ---------------------------------------------------------------------------

typedef float v4f __attribute__((ext_vector_type(4)));
// Match the builtin's parameter type exactly (clang printed
// '__vector_size__(4*sizeof(int)) int __device__ *', i.e. int4 in AS1):
typedef int v4i_vs __attribute__((__vector_size__(4 * sizeof(int))));
typedef __attribute__((address_space(1))) v4i_vs gv4i;   // global (AS1)
typedef __attribute__((address_space(3))) v4i_vs lv4i;   // LDS    (AS3)

#define IMG_W      224u
#define ROW_Q      56u          // 224 / 4 float4's per row
#define PLANE_ELTS (224u * 224u)
#define CEN_LO     80u          // center pixel window [80,160)
#define CEN_HI     160u
#define CEN_QLO    20u          // float4-column window [20,40)
#define CEN_QHI    40u

#if __has_builtin(__builtin_amdgcn_global_load_async_to_lds_b128) && \
    __has_builtin(__builtin_amdgcn_s_wait_asynccnt)
#define USE_ASYNC_LDS 1
#endif

// ---------------------------------------------------------------------------
// Kernel 1: streaming identity copy of everything outside the center window.
// One float4 per thread, non-temporal load+store (stream > 192MB L2).
// ---------------------------------------------------------------------------
__global__ void __launch_bounds__(256) bulk_copy_kernel(
    const float* __restrict__ in, float* __restrict__ out, unsigned n4) {
  unsigned idx = blockIdx.x * 256u + threadIdx.x;
  if (idx >= n4) return;

  unsigned q = idx % ROW_Q;                 // float4 column within row
  unsigned r = (idx / ROW_Q) % IMG_W;       // pixel row within plane

  // Center 5x5 patch block handled by the transpose kernel.
  if (r >= CEN_LO && r < CEN_HI && q >= CEN_QLO && q < CEN_QHI) return;

  const v4f* in4 = (const v4f*)in;
  v4f*       out4 = (v4f*)out;
  v4f v = __builtin_nontemporal_load(in4 + idx);
  __builtin_nontemporal_store(v, out4 + idx);
}

// ---------------------------------------------------------------------------
// Kernel 2: center patches (25 per plane x 768 planes = 19200 patches).
// 4 patches per 256-thread block; 64 threads per 16x16 patch.
//   load : thread t -> row t/4, float4-col t%4  (b128 coalesced,
//          async global->LDS when the gfx1250 builtin exists)
//   LDS  : tile[16][20] floats; stride 20 dwords = 80B keeps every float4
//          slot 16B-aligned and makes the column gather bank-conflict-free
//          (20*c mod 64 distinct for c=0..15).
//   store: thread t -> output row t/4, cols 4*(t%4)..+3, gathering the
//          transposed column from LDS, NT b128 store.
// ---------------------------------------------------------------------------
__global__ void __launch_bounds__(256) center_transpose_kernel(
    const float* __restrict__ in, float* __restrict__ out, unsigned npatch) {
  __shared__ __align__(16) float tile[4][16][20];

  const unsigned lp  = threadIdx.x >> 6;            // local patch 0..3
  const unsigned t   = threadIdx.x & 63u;           // lane within patch
  const unsigned pid = blockIdx.x * 4u + lp;        // global patch id
  const bool active  = pid < npatch;

  const unsigned plane = pid / 25u;
  const unsigned pp    = pid % 25u;
  const unsigned R0    = CEN_LO + (pp / 5u) * 16u;  // patch origin row
  const unsigned C0    = CEN_LO + (pp % 5u) * 16u;  // patch origin col
  const size_t   base  = (size_t)plane * PLANE_ELTS;

  const unsigned row = t >> 2;                      // 0..15
  const unsigned qc  = t & 3u;                      // 0..3 (float4 col)

  if (active) {
    const float* src = in + base + (size_t)(R0 + row) * IMG_W + C0 + qc * 4u;
    float* ldst = &tile[lp][row][qc * 4u];
#ifdef USE_ASYNC_LDS
    // CDNA5 async global->LDS copy, tracked by ASYNCcnt (no VGPR staging).
    __builtin_amdgcn_global_load_async_to_lds_b128(
        (gv4i*)src, (lv4i*)ldst, /*imm offset*/ 0, /*imm cpol*/ 0);
#else
    v4f v = __builtin_nontemporal_load((const v4f*)src);
    *(v4f*)ldst = v;
#endif
  }

#ifdef USE_ASYNC_LDS
  __builtin_amdgcn_s_wait_asynccnt(0);
#endif
  __syncthreads();   // gfx1250: s_barrier_signal -1 / s_barrier_wait -1

  if (active) {
    // out(R0+a3, C0+a4) = x(R0+a4, C0+a3) = tile[a4][a3]
    const unsigned a3 = row;
    v4f v;
    v.x = tile[lp][qc * 4u + 0u][a3];
    v.y = tile[lp][qc * 4u + 1u][a3];
    v.z = tile[lp][qc * 4u + 2u][a3];
    v.w = tile[lp][qc * 4u + 3u][a3];
    float* dst = out + base + (size_t)(R0 + a3) * IMG_W + C0 + qc * 4u;
    __builtin_nontemporal_store(v, (v4f*)dst);
  }
}

// ---------------------------------------------------------------------------
extern "C" void kernel_launch(void* const* d_in, const int* in_sizes, int n_in,
                              void* d_out, int out_size, void* d_ws, size_t ws_size,
                              hipStream_t stream) {
  const float* x = (const float*)d_in[0];
  float* out     = (float*)d_out;

  const unsigned n      = (unsigned)in_sizes[0];        // 256*3*224*224
  const unsigned n4     = n / 4u;                       // float4 count
  const unsigned planes = n / PLANE_ELTS;               // B*C = 768
  const unsigned npatch = planes * 25u;                 // 19200 center patches

  dim3 blk(256);
  bulk_copy_kernel<<<dim3((n4 + 255u) / 256u), blk, 0, stream>>>(x, out, n4);
  center_transpose_kernel<<<dim3((npatch + 3u) / 4u), blk, 0, stream>>>(x, out, npatch);
}